// SBFTransformerConv_80135499809053
// MI455X (gfx1250) — compile-verified
//
#include <hip/hip_runtime.h>
#include <hip/hip_bf16.h>

#define NN 100000
#define EE 800000
#define HH 4
#define CC 32

typedef float v2f __attribute__((ext_vector_type(2)));
typedef float v8f __attribute__((ext_vector_type(8)));

__device__ __forceinline__ v8f wmma4(v2f a, v2f b, v8f c) {
  // D(16x16,f32) = A(16x4,f32) x B(4x16,f32) + C
  return __builtin_amdgcn_wmma_f32_16x16x4_f32(false, a, false, b, (short)0, c, false, false);
}

// One 16x16 D-tile of A(16xK, LDS row-major lda) @ B(KxN global row-major ldb, this
// tile's 16 columns start at B). K must be a multiple of 4; B row loads are guarded
// against Kvalid (A must be zero-padded past Kvalid).
__device__ __forceinline__ v8f gemm_tile(const float* A, int lda,
                                         const float* __restrict__ B, int ldb,
                                         int K, int Kvalid, v8f acc) {
  const int lane = threadIdx.x & 31;
  const int half = lane >> 4;
  const int mn = lane & 15;
  #pragma unroll
  for (int k = 0; k < K; k += 4) {
    const int k0 = k + 2 * half;
    v2f a, b;
    a[0] = A[mn * lda + k0];
    a[1] = A[mn * lda + k0 + 1];
    b[0] = (k0     < Kvalid) ? B[(size_t)(k0    ) * ldb + mn] : 0.f;
    b[1] = (k0 + 1 < Kvalid) ? B[(size_t)(k0 + 1) * ldb + mn] : 0.f;
    acc = wmma4(a, b, acc);
  }
  return acc;
}

// Scatter a D-tile (v8f accumulator layout) into LDS: row = r + 8*half, col = mn.
__device__ __forceinline__ void store_tile_lds(float* dst, int ldd, v8f acc) {
  const int lane = threadIdx.x & 31;
  const int half = lane >> 4;
  const int mn = lane & 15;
  #pragma unroll
  for (int r = 0; r < 8; ++r)
    dst[(r + 8 * half) * ldd + mn] = acc[r];
}

// [16,32] (LDS, lda=33) @ B[32,128] + bias -> global out rows n0..n0+15 (ld 128)
__device__ __forceinline__ void gemm_store_128(const float (*A)[33],
                                               const float* __restrict__ B,
                                               const float* __restrict__ bias,
                                               float* __restrict__ out, int n0) {
  const int lane = threadIdx.x & 31;
  const int half = lane >> 4;
  const int mn = lane & 15;
  #pragma unroll
  for (int nt = 0; nt < 8; ++nt) {
    v8f acc = {};
    acc = gemm_tile(&A[0][0], 33, B + nt * 16, 128, 32, 32, acc);
    const float bv = bias[nt * 16 + mn];
    #pragma unroll
    for (int r = 0; r < 8; ++r)
      out[(size_t)(n0 + r + 8 * half) * 128 + nt * 16 + mn] = acc[r] + bv;
  }
}

__device__ __forceinline__ void atomicMaxF(float* addr, float val) {
  if (val >= 0.f) atomicMax((int*)addr, __float_as_int(val));
  else            atomicMin((unsigned int*)addr, (unsigned int)__float_as_int(val));
}

// ---------------------------------------------------------------- kernels ----

__global__ __launch_bounds__(256) void init_kernel(float* __restrict__ amax,
                                                   float* __restrict__ denom) {
  const int gid = blockIdx.x * 256 + threadIdx.x;
  if (gid < NN * HH) {
    amax[gid] = -__builtin_inff();
    denom[gid] = 0.f;
  }
}

// One wave handles 16 nodes: rbf filter chain + K/Q/V/skip projections, all WMMA fp32.
__global__ __launch_bounds__(32) void node_kernel(
    const float* __restrict__ x, const float* __restrict__ rbf,
    const float* __restrict__ w_rbf0, const float* __restrict__ w_rbf1,
    const float* __restrict__ w_k, const float* __restrict__ b_k,
    const float* __restrict__ w_q, const float* __restrict__ b_q,
    const float* __restrict__ w_v, const float* __restrict__ b_v,
    const float* __restrict__ w_skip, const float* __restrict__ b_skip,
    float* __restrict__ key, float* __restrict__ value,
    float* __restrict__ query, float* __restrict__ out) {
  __shared__ float sRBF[16][45];  // 16x42 zero-padded to K=44
  __shared__ float sX[16][33];
  __shared__ float sU[16][33];    // rbf @ w_rbf0
  __shared__ float sXS[16][33];   // x_src = rbf_filter * x
  const int lane = threadIdx.x;
  const int n0 = blockIdx.x * 16;

  for (int i = lane; i < 16 * 45; i += 32) {
    const int r = i / 45, c = i % 45;
    sRBF[r][c] = (c < 42) ? rbf[(size_t)(n0 + r) * 42 + c] : 0.f;
  }
  for (int i = lane; i < 16 * 32; i += 32) {
    const int r = i >> 5, c = i & 31;
    sX[r][c] = x[(size_t)(n0 + r) * 32 + c];
  }
  __syncthreads();

  // tmp = rbf @ w_rbf0   (K padded 42->44, guard B rows)
  #pragma unroll
  for (int nt = 0; nt < 2; ++nt) {
    v8f acc = {};
    acc = gemm_tile(&sRBF[0][0], 45, w_rbf0 + nt * 16, 32, 44, 42, acc);
    store_tile_lds(&sU[0][nt * 16], 33, acc);
  }
  __syncthreads();

  // rbf_filter = tmp @ w_rbf1 ; x_src = rbf_filter * x
  {
    const int half = lane >> 4, mn = lane & 15;
    #pragma unroll
    for (int nt = 0; nt < 2; ++nt) {
      v8f acc = {};
      acc = gemm_tile(&sU[0][0], 33, w_rbf1 + nt * 16, 32, 32, 32, acc);
      #pragma unroll
      for (int r = 0; r < 8; ++r) {
        const int row = r + 8 * half, col = nt * 16 + mn;
        sXS[row][col] = acc[r] * sX[row][col];
      }
    }
  }
  __syncthreads();

  gemm_store_128(sXS, w_k, b_k, key, n0);
  gemm_store_128(sXS, w_v, b_v, value, n0);
  gemm_store_128(sX, w_q, b_q, query, n0);
  gemm_store_128(sX, w_skip, b_skip, out, n0);  // root skip written directly to output
}

// alpha_raw[e,h] = (q[dst,h] . (k[src,h] + edge_attr[e]@w_ek[:,h])) / sqrt(C); segment max
__global__ __launch_bounds__(256) void edge_alpha_kernel(
    const float* __restrict__ edge_attr, const float* __restrict__ w_ek,
    const float* __restrict__ key, const float* __restrict__ query,
    const int* __restrict__ edge_index,
    float* __restrict__ alpha, float* __restrict__ amax) {
  const int gid = blockIdx.x * 256 + threadIdx.x;
  if (gid >= EE * HH) return;
  const int e = gid >> 2, h = gid & 3;
  const int src = edge_index[e], dst = edge_index[EE + e];
  float ea[16];
  #pragma unroll
  for (int d = 0; d < 16; ++d) ea[d] = edge_attr[(size_t)e * 16 + d];
  const float* kp = key + (size_t)src * 128 + h * 32;
  const float* qp = query + (size_t)dst * 128 + h * 32;
  const float* wp = w_ek + h * 32;
  float acc = 0.f;
  #pragma unroll 4
  for (int c = 0; c < 32; ++c) {
    float ekc = 0.f;
    #pragma unroll
    for (int d = 0; d < 16; ++d) ekc = fmaf(ea[d], wp[(size_t)d * 128 + c], ekc);
    acc = fmaf(qp[c], kp[c] + ekc, acc);
  }
  acc *= 0.17677669529663687f;  // 1/sqrt(32)
  alpha[gid] = acc;
  atomicMaxF(&amax[dst * 4 + h], acc);
}

// ex = exp(alpha - amax[dst]); segment sum into denom
__global__ __launch_bounds__(256) void edge_exp_kernel(
    const int* __restrict__ edge_index, float* __restrict__ alpha,
    const float* __restrict__ amax, float* __restrict__ denom) {
  const int gid = blockIdx.x * 256 + threadIdx.x;
  if (gid >= EE * HH) return;
  const int e = gid >> 2, h = gid & 3;
  const int dst = edge_index[EE + e];
  float m = amax[dst * 4 + h];
  if (m < -1e37f) m = 0.f;  // isfinite() guard (amax only ever -inf or finite)
  const float ex = __expf(alpha[gid] - m);
  alpha[gid] = ex;
  atomicAdd(&denom[dst * 4 + h], ex);
}

// One wave handles 16 edges: edge_val + sbf_filter via WMMA, then message scatter.
__global__ __launch_bounds__(32) void edge_message_kernel(
    const float* __restrict__ edge_attr, const float* __restrict__ sbf,
    const float* __restrict__ w_ev, const float* __restrict__ w_sbf0,
    const float* __restrict__ w_sbf1, const float* __restrict__ value,
    const int* __restrict__ edge_index, const float* __restrict__ alpha,
    const float* __restrict__ denom, float* __restrict__ out) {
  __shared__ float sEA[16][17];    // edge_attr tile
  __shared__ float sSBF[64][17];   // sbf rows (e,h)
  __shared__ float sTMP[16][33];   // sbf @ w_sbf0
  __shared__ float sSF[64][33];    // sbf_filter
  __shared__ float sEV[16][129];   // edge_val
  const int lane = threadIdx.x;
  const int e0 = blockIdx.x * 16;

  for (int i = lane; i < 16 * 16; i += 32) {
    const int r = i >> 4, c = i & 15;
    sEA[r][c] = edge_attr[(size_t)(e0 + r) * 16 + c];
  }
  for (int i = lane; i < 64 * 16; i += 32) {
    const int r = i >> 4, c = i & 15;
    sSBF[r][c] = sbf[(size_t)(e0 * 4 + r) * 16 + c];
  }
  __syncthreads();

  // edge_val[16,128] = edge_attr @ w_ev
  #pragma unroll
  for (int nt = 0; nt < 8; ++nt) {
    v8f acc = {};
    acc = gemm_tile(&sEA[0][0], 17, w_ev + nt * 16, 128, 16, 16, acc);
    store_tile_lds(&sEV[0][nt * 16], 129, acc);
  }

  // sbf_filter[64,32] = (sbf @ w_sbf0) @ w_sbf1, 4 row-tiles of 16
  for (int rt = 0; rt < 4; ++rt) {
    #pragma unroll
    for (int nt = 0; nt < 2; ++nt) {
      v8f acc = {};
      acc = gemm_tile(&sSBF[rt * 16][0], 17, w_sbf0 + nt * 16, 32, 16, 16, acc);
      store_tile_lds(&sTMP[0][nt * 16], 33, acc);
    }
    __syncthreads();
    #pragma unroll
    for (int nt = 0; nt < 2; ++nt) {
      v8f acc = {};
      acc = gemm_tile(&sTMP[0][0], 33, w_sbf1 + nt * 16, 32, 32, 32, acc);
      store_tile_lds(&sSF[rt * 16][nt * 16], 33, acc);
    }
    __syncthreads();
  }

  // message = (value[src] + edge_val) * (ex/denom[dst]) * sbf_filter ; scatter-add
  for (int el = 0; el < 16; ++el) {
    const int e = e0 + el;
    const int src = edge_index[e], dst = edge_index[EE + e];
    #pragma unroll
    for (int q = 0; q < 4; ++q) {
      const int col = q * 32 + lane;  // h = q, c = lane
      const float aex = alpha[e * 4 + q];
      const float den = denom[dst * 4 + q];
      const float a = aex / (den + 1e-16f);
      const float msg =
          (value[(size_t)src * 128 + col] + sEV[el][col]) * a * sSF[el * 4 + q][lane];
      atomicAdd(&out[(size_t)dst * 128 + col], msg);
    }
  }
}

// ------------------------------------------------------------------ launch ----

extern "C" void kernel_launch(void* const* d_in, const int* in_sizes, int n_in,
                              void* d_out, int out_size, void* d_ws, size_t ws_size,
                              hipStream_t stream) {
  const float* x         = (const float*)d_in[0];
  const float* edge_attr = (const float*)d_in[1];
  const float* rbf       = (const float*)d_in[2];
  const float* sbf       = (const float*)d_in[3];
  const float* w_rbf0    = (const float*)d_in[4];
  const float* w_rbf1    = (const float*)d_in[5];
  const float* w_sbf0    = (const float*)d_in[6];
  const float* w_sbf1    = (const float*)d_in[7];
  const float* w_ek      = (const float*)d_in[8];
  const float* w_ev      = (const float*)d_in[9];
  const float* w_k       = (const float*)d_in[10];
  const float* b_k       = (const float*)d_in[11];
  const float* w_q       = (const float*)d_in[12];
  const float* b_q       = (const float*)d_in[13];
  const float* w_v       = (const float*)d_in[14];
  const float* b_v       = (const float*)d_in[15];
  const float* w_skip    = (const float*)d_in[16];
  const float* b_skip    = (const float*)d_in[17];
  const int* edge_index  = (const int*)d_in[18];
  float* out = (float*)d_out;

  float* ws    = (float*)d_ws;
  float* key   = ws;                         // N*128
  float* value = key + (size_t)NN * 128;     // N*128
  float* query = value + (size_t)NN * 128;   // N*128
  float* alpha = query + (size_t)NN * 128;   // E*H
  float* amax  = alpha + (size_t)EE * HH;    // N*H
  float* denom = amax + (size_t)NN * HH;     // N*H

  init_kernel<<<(NN * HH + 255) / 256, 256, 0, stream>>>(amax, denom);

  node_kernel<<<NN / 16, 32, 0, stream>>>(x, rbf, w_rbf0, w_rbf1, w_k, b_k, w_q, b_q,
                                          w_v, b_v, w_skip, b_skip, key, value, query,
                                          out);

  edge_alpha_kernel<<<(EE * HH) / 256, 256, 0, stream>>>(edge_attr, w_ek, key, query,
                                                         edge_index, alpha, amax);

  edge_exp_kernel<<<(EE * HH) / 256, 256, 0, stream>>>(edge_index, alpha, amax, denom);

  edge_message_kernel<<<EE / 16, 32, 0, stream>>>(edge_attr, sbf, w_ev, w_sbf0, w_sbf1,
                                                  value, edge_index, alpha, denom, out);
}